// GTMapBoundLoss_37108517438044
// MI455X (gfx1250) — compile-verified
//
#include <hip/hip_runtime.h>
#include <math.h>

typedef __attribute__((ext_vector_type(2))) float v2f;
typedef __attribute__((ext_vector_type(8))) float v8f;

#define B_ 64
#define T_ 12
#define N_ 2000
#define P_ 20
#define NP (N_*P_)         // 40000 points per batch
#define CLS_IDX_ 2
#define BIGD2 1.0e18f      // squared-distance sentinel (sqrt -> 1e9 == reference BIG)

// One workgroup (8 wave32) per batch element.
__global__ __launch_bounds__(256)
void gtmap_loss_main(const float* __restrict__ pred,    // (B,T,2)
                     const float* __restrict__ pts,     // (B,N,1,P,2)
                     const int*   __restrict__ labels,  // (B,N)
                     const float* __restrict__ masks,   // (B,T)
                     float* __restrict__ ws_total,      // (B)
                     float* __restrict__ ws_count)      // (B)
{
    __shared__ int   sh_vlist[N_];      // compacted indices of valid lines
    __shared__ int   sh_vcnt;
    __shared__ float sh_pred[T_*2];
    __shared__ float sh_s1[T_*2];
    __shared__ float sh_d1[T_*2];
    __shared__ float sh_min[8][16];     // per-wave min dist^2 per t
    __shared__ unsigned sh_cross;

    const int b    = blockIdx.x;
    const int tid  = threadIdx.x;
    const int lane = tid & 31;
    const int wave = tid >> 5;
    const bool half0 = lane < 16;
    const int row  = lane & 15;

    const float* ptsb = pts + (size_t)b * NP * 2;

    if (tid == 0) { sh_vcnt = 0; sh_cross = 0u; }
    __syncthreads();

    // ---- warm WGP$/L2: 320 KB of line points = 2500 cachelines ----
    {
        const char* pc = (const char*)ptsb;
        for (int i = tid; i < (NP*8)/128; i += 256)
            __builtin_prefetch(pc + i*128, 0, 3);
    }
    // ---- compact valid-line indices into LDS (order-independent consumers) ----
    for (int i = tid; i < N_; i += 256) {
        if (labels[b*N_ + i] == CLS_IDX_) {
            int p = atomicAdd(&sh_vcnt, 1);
            sh_vlist[p] = i;
        }
    }
    // ---- preds + trajectory segments (origin-prepended) ----
    if (tid < T_) {
        float px = pred[(b*T_ + tid)*2 + 0];
        float py = pred[(b*T_ + tid)*2 + 1];
        sh_pred[tid*2+0] = px; sh_pred[tid*2+1] = py;
        float sx = 0.f, sy = 0.f;
        if (tid > 0) {
            sx = pred[(b*T_ + tid-1)*2 + 0];
            sy = pred[(b*T_ + tid-1)*2 + 1];
        }
        sh_s1[tid*2+0] = sx;      sh_s1[tid*2+1] = sy;
        sh_d1[tid*2+0] = px - sx; sh_d1[tid*2+1] = py - sy;
    }
    __syncthreads();

    const int V  = sh_vcnt;        // number of valid lines (~N/4)
    const int VP = V * P_;         // number of valid points

    // ============ Pass 1: min squared distance via V_WMMA_F32_16X16X4_F32 ============
    // dist^2(m,n) = |p_m|^2 - 2 p_m.q_n + |q_n|^2
    // A (16x4): row m = (px, py, 1, 0)   [loop invariant]
    // B (4x16): col n = (-2qx, -2qy, qq, 0)
    // C (16x16): row m = |p_m|^2         [loop invariant]
    float apx = (row < T_) ? sh_pred[row*2+0] : 0.f;
    float apy = (row < T_) ? sh_pred[row*2+1] : 0.f;
    v2f av;
    av.x = half0 ? apx : 1.0f;   // K0 | K2
    av.y = half0 ? apy : 0.0f;   // K1 | K3
    v8f cvec;
    #pragma unroll
    for (int r = 0; r < 8; ++r) {
        int m = half0 ? r : (r + 8);
        float ppx = (m < T_) ? sh_pred[m*2+0] : 0.f;
        float ppy = (m < T_) ? sh_pred[m*2+1] : 0.f;
        cvec[r] = ppx*ppx + ppy*ppy;
    }
    v8f m8;
    #pragma unroll
    for (int r = 0; r < 8; ++r) m8[r] = 3.0e38f;

    const int ntile = (VP + 15) >> 4;          // uniform across the block
    for (int tile = wave; tile < ntile; tile += 8) {
        int gv  = tile*16 + row;               // dense valid-point index
        bool inr = gv < VP;                    // tail-pad lanes (last tile only)
        int gvc = inr ? gv : 0;
        int li  = gvc / P_;                    // index into sh_vlist
        int pj  = gvc - li*P_;
        int line = sh_vlist[li];
        v2f q = *(const v2f*)(ptsb + (line*P_ + pj)*2);
        float qq = inr ? (q.x*q.x + q.y*q.y) : BIGD2;
        v2f bv;
        bv.x = half0 ? (-2.0f*q.x) : qq;       // K0 | K2
        bv.y = half0 ? (-2.0f*q.y) : 0.0f;     // K1 | K3
        v8f d = __builtin_amdgcn_wmma_f32_16x16x4_f32(
            false, av, false, bv, (short)0, cvec, false, false);
        #pragma unroll
        for (int r = 0; r < 8; ++r) m8[r] = fminf(m8[r], d[r]);
    }
    // reduce over the 16 N-columns held within each 16-lane half (wave32 butterfly)
    #pragma unroll
    for (int r = 0; r < 8; ++r) {
        float v = m8[r];
        v = fminf(v, __shfl_xor(v, 8, 32));
        v = fminf(v, __shfl_xor(v, 4, 32));
        v = fminf(v, __shfl_xor(v, 2, 32));
        v = fminf(v, __shfl_xor(v, 1, 32));
        m8[r] = v;
    }
    if (lane == 0) {                 // lanes 0-15 hold rows M=0..7
        #pragma unroll
        for (int r = 0; r < 8; ++r) sh_min[wave][r] = m8[r];
    }
    if (lane == 16) {                // lanes 16-31 hold rows M=8..15
        #pragma unroll
        for (int r = 0; r < 8; ++r) sh_min[wave][8+r] = m8[r];
    }

    // ============ Pass 2: segment intersections, dense over valid lines ============
    float s1x[T_], s1y[T_], d1x[T_], d1y[T_];
    #pragma unroll
    for (int t = 0; t < T_; ++t) {
        s1x[t] = sh_s1[t*2+0]; s1y[t] = sh_s1[t*2+1];
        d1x[t] = sh_d1[t*2+0]; d1y[t] = sh_d1[t*2+1];
    }
    unsigned cross = 0u;
    const int nseg = V * (P_ - 1);
    for (int s = tid; s < nseg; s += 256) {
        int li = s / (P_-1);
        int j  = s - li*(P_-1);
        int line = sh_vlist[li];
        const float* lp = ptsb + (line*P_ + j)*2;
        float bsx = lp[0], bsy = lp[1], bex = lp[2], bey = lp[3];
        float d2x = bex - bsx, d2y = bey - bsy;
        #pragma unroll
        for (int t = 0; t < T_; ++t) {
            float det = d1x[t]*d2y - d2x*d1y[t];
            if (det != 0.f) {
                float rx = bsx - s1x[t], ry = bsy - s1y[t];
                float t1 = (rx*d2y - ry*d2x) / det;
                float t2 = (rx*d1y[t] - ry*d1x[t]) / det;
                if (t1 >= 0.f && t1 <= 1.f && t2 >= 0.f && t2 <= 1.f)
                    cross |= (1u << t);
            }
        }
    }
    if (cross) atomicOr(&sh_cross, cross);
    __syncthreads();

    // ============ Per-batch mask / cumsum logic ============
    if (tid == 0) {
        bool hasl = (V > 0);
        unsigned cr = sh_cross;
        float total = 0.f, cnt = 0.f;
        bool blocked = false;
        for (int t = 0; t < T_; ++t) {
            float md2 = sh_min[0][t];
            for (int w = 1; w < 8; ++w) md2 = fminf(md2, sh_min[w][t]);
            md2 = fmaxf(md2, 0.f);                       // guard tiny negative from fma
            float lt = fmaxf(0.f, 1.0f - sqrtf(md2));    // relu(DIS_THRESH - min_dist)
            bool validt = (masks[b*T_ + t] >= 0.5f) && hasl;
            bool crt = ((cr >> t) & 1u) != 0u;
            blocked = blocked || (validt && crt);        // cumsum(trigger) > 0 (inclusive)
            if (validt && !blocked) { total += lt; cnt += 1.f; }
        }
        ws_total[b] = total;
        ws_count[b] = cnt;
    }
}

// Deterministic in-order final reduction over the 64 batch partials.
__global__ void gtmap_loss_final(const float* __restrict__ ws_total,
                                 const float* __restrict__ ws_count,
                                 float* __restrict__ out)
{
    if (threadIdx.x == 0 && blockIdx.x == 0) {
        float total = 0.f, n = 0.f;
        for (int b2 = 0; b2 < B_; ++b2) { total += ws_total[b2]; n += ws_count[b2]; }
        out[0] = (n == 0.f) ? 0.f : (total / fmaxf(n, 1.0f));  // WEIGHT == 1.0
    }
}

extern "C" void kernel_launch(void* const* d_in, const int* in_sizes, int n_in,
                              void* d_out, int out_size, void* d_ws, size_t ws_size,
                              hipStream_t stream)
{
    const float* pred   = (const float*)d_in[0];  // (B,T,2)  f32
    const float* pts    = (const float*)d_in[1];  // (B,N,1,P,2) f32
    const int*   labels = (const int*)  d_in[2];  // (B,N)  i32
    const float* masks  = (const float*)d_in[3];  // (B,T)  f32
    float* out = (float*)d_out;

    float* ws_total = (float*)d_ws;       // 64 floats
    float* ws_count = ws_total + B_;      // 64 floats

    gtmap_loss_main<<<B_, 256, 0, stream>>>(pred, pts, labels, masks, ws_total, ws_count);
    gtmap_loss_final<<<1, 32, 0, stream>>>(ws_total, ws_count, out);
}